// SO2onS2_51737176048547
// MI455X (gfx1250) — compile-verified
//
#include <hip/hip_runtime.h>
#include <hip/hip_bf16.h>

typedef float v2f __attribute__((ext_vector_type(2)));
typedef float v8f __attribute__((ext_vector_type(8)));

#define NPAIRS 21
#define NOUT   69

// Pair tables (l1<=l2<=5), in reference PAIRS order.
static constexpr int PL1[NPAIRS]  = {0,0,0,0,0,0, 1,1,1,1,1, 2,2,2,2, 3,3,3, 4,4, 5};
static constexpr int PL2[NPAIRS]  = {0,1,2,3,4,5, 1,2,3,4,5, 2,3,4,5, 3,4,5, 4,5, 5};
static constexpr int PD [NPAIRS]  = {1,3,5,7,9,11, 9,15,21,27,33, 25,35,45,55, 49,63,77, 81,99, 121};
static constexpr int POFF[NPAIRS] = {0,1,10,35,84,165, 286,367,592,1033,1762,
                                     2851,3476,4701,6726, 9751,12152,16121, 22050,28611, 38412};
static constexpr int PNL[NPAIRS]  = {1,1,1,1,1,1, 3,3,3,3,2, 5,5,4,3, 6,5,4, 6,5, 6};
static constexpr int PLMIN[NPAIRS]= {0,1,2,3,4,5, 0,1,2,3,4, 0,1,2,3, 0,1,2, 0,1, 0};
static constexpr int POB[NPAIRS]  = {0,1,2,3,4,5, 6,9,12,15,18, 20,25,30,34, 37,43,48, 52,58, 63};
// Base (in floats) of each pair's fragment-ordered CG block: sum of dpad*16, dpad=ceil(d/4)*4
static constexpr int WBASE[NPAIRS]= {0,64,128,256,384,576, 768,960,1216,1600,2048,
                                     2624,3072,3648,4416, 5312,6144,7168, 8448,9792, 11392};
#define WTOT 13376   // total fragment floats (53.5 KB)

// 8 waves per block, each wave owns a 16-row batch tile.
__global__ __launch_bounds__(256)
void bispectrum_wmma_kernel(const float* __restrict__ cr,
                            const float* __restrict__ ci,
                            const float* __restrict__ cg,
                            float* __restrict__ out,
                            int nB)
{
    // CG staged in WMMA-B fragment order: Wfrag[wbase + t*64 + n*4 + K], k = 4t+K.
    __shared__ float  Wfrag[WTOT];
    // f[l][m] complex interleaved; row padded to 37 so dead-lane reads stay in-bounds.
    __shared__ float2 fcx[8][16][37];

    const int tid  = threadIdx.x;
    const int wave = tid >> 5;
    const int lane = tid & 31;
    const int b0   = (blockIdx.x * 8 + wave) * 16;
    const bool active = (b0 < nB);          // wave-uniform

    // ---- Phase 0: stage CG fragments into LDS via async gather->LDS-scatter ----
#pragma unroll
    for (int p = 0; p < NPAIRS; ++p) {
        const int d    = PD[p];
        const int nl   = PNL[p];
        const int lmin = PLMIN[p];
        const int dp16 = (((d + 3) >> 2) << 2) * 16;
        const long cgoff = (long)POFF[p];
        for (int e = tid; e < dp16; e += 256) {
            const int K = e & 3;
            const int n = (e >> 2) & 15;
            const int t = e >> 6;
            const int k = 4 * t + K;
            const unsigned ldsoff =
                (unsigned)(size_t)&Wfrag[WBASE[p] + e];   // low 32 bits = LDS offset
            if (k < d && n < nl) {
                const int col = lmin * (2 * n + 1) + n * (n + 1);
                const unsigned long long gaddr =
                    (unsigned long long)(size_t)(cg + cgoff + (long)k * d + col);
                asm volatile("global_load_async_to_lds_b32 %0, %1, off"
                             :: "v"(ldsoff), "v"(gaddr) : "memory");
            } else {
                Wfrag[WBASE[p] + e] = 0.0f;               // zero padding (plain DS store)
            }
        }
    }

    // ---- Phase 1: build f tables (lanes 0..15 each own one batch row) ----
    if (active && lane < 16) {
        const size_t b = (size_t)(b0 + lane);
        const float* crb = cr + b * 36;
        const float* cib = ci + b * 36;
#pragma unroll
        for (int l = 0; l < 6; ++l) {
#pragma unroll
            for (int m = 0; m <= l; ++m) {
                float pr = crb[l * 6 + m];
                float pi = cib[l * 6 + m];
                float2 v; v.x = pr; v.y = pi;
                fcx[wave][lane][l*l + l + m] = v;
                if (m > 0) {                       // f(-m) = (-1)^m conj(f(m))
                    float s = (m & 1) ? -1.0f : 1.0f;
                    float2 w; w.x = s * pr; w.y = -s * pi;
                    fcx[wave][lane][l*l + l - m] = w;
                }
            }
        }
    }
    asm volatile("s_wait_asynccnt 0" ::: "memory");       // async scatters landed
    __syncthreads();

    if (!active) return;                    // wave-uniform: EXEC stays all-1 inside

    const int m15  = lane & 15;             // A-row (batch) AND B-column this lane owns
    const int hi   = lane >> 4;             // lanes >=16 own K = {2,3}
    const int koff = hi * 2;
    const int rbase = hi * 8;               // C rows this lane holds: rbase..rbase+7

#pragma unroll
    for (int p = 0; p < NPAIRS; ++p) {
        const int l2   = PL2[p];
        const int d    = PD[p];
        const int nl   = PNL[p];
        const int lmin = PLMIN[p];
        const int m2   = 2 * l2 + 1;        // constant after unroll -> div folds
        const int f1o  = PL1[p] * PL1[p];
        const int f2o  = l2 * l2;
        const int nsteps = (d + 3) >> 2;
        const float* wp = &Wfrag[WBASE[p] + m15 * 4 + koff];
        const float2* frow = &fcx[wave][m15][0];

        v8f acc_re = {};
        v8f acc_im = {};

        for (int t = 0; t < nsteps; ++t) {
            const int kb = t * 4 + koff;
            float ar[2], ai[2];
#pragma unroll
            for (int u = 0; u < 2; ++u) {
                const int  k  = kb + u;
                const bool ok = (k < d);
                const int  i  = k / m2;              // magic-mul (const divisor)
                const int  j  = k - i * m2;
                // A element: tp[m15,k] = f[l1][m15,i] * f[l2][m15,j]  (2x ds_load_b64)
                float2 F1 = frow[f1o + i];
                float2 F2 = frow[f2o + j];
                ar[u] = ok ? (F1.x * F2.x - F1.y * F2.y) : 0.0f;
                ai[u] = ok ? (F1.x * F2.y + F1.y * F2.x) : 0.0f;
            }
            // B fragment: one aligned ds_load_b64, zero-padded, no predication
            const float2 bw = *(const float2*)(wp + t * 64);
            v2f a_re; a_re.x = ar[0]; a_re.y = ar[1];
            v2f a_im; a_im.x = ai[0]; a_im.y = ai[1];
            v2f bfr;  bfr.x  = bw.x;  bfr.y  = bw.y;

            // D = A x B + C   (16x16x4 f32, full precision)
            acc_re = __builtin_amdgcn_wmma_f32_16x16x4_f32(
                         false, a_re, false, bfr, (short)0, acc_re, false, false);
            acc_im = __builtin_amdgcn_wmma_f32_16x16x4_f32(
                         false, a_im, false, bfr, (short)0, acc_im, false, false);
        }

        // ---- Epilogue: T * conj(f[l][m=0]) for this lane's output column ----
        if (m15 < nl) {
            const int l    = lmin + m15;
            const int fidx = l * l + l;              // m = 0 slot
            const int tcol = POB[p] + m15;           // global output index
#pragma unroll
            for (int v = 0; v < 8; ++v) {
                const int r  = rbase + v;
                float2 F0 = fcx[wave][r][fidx];
                float Tr  = acc_re[v];
                float Ti  = acc_im[v];
                float o_r = Tr * F0.x + Ti * F0.y;   // (Tr+iTi)*(F0r-iF0i)
                float o_i = Ti * F0.x - Tr * F0.y;
                long  o   = ((long)(b0 + r) * NOUT + tcol) * 2;
                float2 val; val.x = o_r; val.y = o_i;
                *(float2*)(out + o) = val;           // 8B store, complex64 interleave
            }
        }
    }
}

extern "C" void kernel_launch(void* const* d_in, const int* in_sizes, int n_in,
                              void* d_out, int out_size, void* d_ws, size_t ws_size,
                              hipStream_t stream)
{
    const float* cr = (const float*)d_in[0];   // coeffs_real (B,6,6)
    const float* ci = (const float*)d_in[1];   // coeffs_imag (B,6,6)
    const float* cg = (const float*)d_in[2];   // cg_flat (53053,)
    float* out = (float*)d_out;                // (B,69) complex64 -> interleaved f32

    const int nB = in_sizes[0] / 36;
    const int rows_per_block = 8 * 16;         // 8 waves x 16-row tile
    const int grid = (nB + rows_per_block - 1) / rows_per_block;

    bispectrum_wmma_kernel<<<grid, 256, 0, stream>>>(cr, ci, cg, out, nB);
}